// CANN_3822520893875
// MI455X (gfx1250) — compile-verified
//
#include <hip/hip_runtime.h>
#include <hip/hip_bf16.h>
#include <math.h>

// CANN ring-attractor, persistent single-kernel formulation for gfx1250.
//  - B=128 rings -> 8 workgroups of 16 batch rows
//  - N=100 neurons padded to NP=112 -> 7 waves of 32 (one 16-col tile per wave)
//  - circulant conv done as chained V_WMMA_F32_16X16X4_F32 (fp32 exact path)
//  - loop-invariant B fragments (circulant matrix) hoisted into 56 VGPRs/lane,
//    so each time step only loads A fragments (r_eff) from LDS
//  - all state held in VGPRs for 256 steps

typedef __attribute__((ext_vector_type(2))) float v2f;
typedef __attribute__((ext_vector_type(8))) float v8f;

#define N_REAL   100
#define NP       112          // padded N: 7 tiles of 16
#define NK4      (NP / 4)     // 28 k-steps of 4
#define NSTEPS   256
#define BROWS    16           // batch rows per workgroup
#define NWAVES   7            // NP / 16
#define BLOCKT   (NWAVES * 32)

#define DT_OVER_TAU  0.01f    // DT/TAU = 0.1/10
#define K_RHO        0.5f     // K * RHO
#define INV_TAU_D    (1.0f / 3.0f)
#define INV_TAU_F    (1.0f / 0.3f)
#define U_STP_C      0.45f
#define DT_SEC       1.0e-4f  // DT/1000

__global__ __launch_bounds__(BLOCKT) void cann_persistent(
    const float* __restrict__ gu, const float* __restrict__ gr,
    const float* __restrict__ gx, const float* __restrict__ gsu,
    const float* __restrict__ gI, const float* __restrict__ gkern,
    float* __restrict__ out)
{
    // LDS: circulant matrix stored K-contiguous: Wnk[n][k] = kernel[(n-k) mod 100]
    __shared__ float Wnk[NP][NP];       // 49.0 KB
    __shared__ float reff[BROWS][NP];   //  7.0 KB
    __shared__ float rowsum[2][BROWS];  // double-buffered norm accumulators

    const int tid  = threadIdx.x;
    const int wave = tid >> 5;          // 0..6 -> owns output cols [16w,16w+16)
    const int lane = tid & 31;
    const int half = lane >> 4;         // 0 | 1
    const int l16  = lane & 15;
    const int bx   = blockIdx.x;        // 0..7
    const int col  = 16 * wave + l16;   // this lane's neuron column (C/D layout N)
    const int rowbase = bx * BROWS;

    // ---- build circulant matrix in LDS (once) ----
    for (int i = tid; i < NP * NP; i += BLOCKT) {
        int n = i / NP, k = i % NP;
        float v = 0.0f;
        if (n < N_REAL && k < N_REAL) {
            int d = n - k;                 // in (-100, 100)
            d = (d + N_REAL) % N_REAL;
            v = gkern[d];
        }
        Wnk[n][k] = v;
    }
    if (tid < 2 * BROWS) rowsum[tid >> 4][tid & 15] = 0.0f;

    // ---- load state into registers, WMMA C/D fragment layout ----
    // element v of this lane = (row m = v + 8*half, neuron col)
    float u[8], r[8], x[8], su[8], Ie[8], usq[8];
    const bool valid = (col < N_REAL);
#pragma unroll
    for (int v = 0; v < 8; ++v) {
        int m = v + 8 * half;
        int g = (rowbase + m) * N_REAL + col;
        u[v]  = valid ? gu[g]  : 0.0f;
        r[v]  = valid ? gr[g]  : 0.0f;
        x[v]  = valid ? gx[g]  : 1.0f;
        su[v] = valid ? gsu[g] : U_STP_C;
        Ie[v] = valid ? gI[g]  : 0.0f;
    }
    __syncthreads();

    // ---- hoist loop-invariant B fragments into registers ----
    // B-fragment element (K, N): lane holds N = col, K = 4*kk + 2*half + {0,1}
    const float* brow = &Wnk[col][2 * half];
    v2f bfrag[NK4];
#pragma unroll
    for (int kk = 0; kk < NK4; ++kk)
        bfrag[kk] = *(const v2f*)(brow + 4 * kk);

    // A-fragment source: r_eff[M = l16][K = 4*kk + 2*half + {0,1}]
    const float* arow = &reff[l16][2 * half];

    // ---- time-step recurrence ----
    for (int s = 0; s < NSTEPS; ++s) {
        // 1) r_eff = r * su * x  -> LDS (C-layout scatter: 8 rows at this col)
#pragma unroll
        for (int v = 0; v < 8; ++v)
            reff[8 * half + v][col] = r[v] * su[v] * x[v];
        __syncthreads();                              // barrier A

        // zero the buffer the NEXT step accumulates into (reads of it done pre-A)
        if (wave == 0 && lane < BROWS) rowsum[(s + 1) & 1][lane] = 0.0f;

        // 2) recurrent = r_eff @ Wnk : 28 chained fp32 WMMAs (A from LDS only)
        v8f acc = {};
#pragma unroll
        for (int kk = 0; kk < NK4; ++kk) {
            v2f a = *(const v2f*)(arow + 4 * kk);
            acc = __builtin_amdgcn_wmma_f32_16x16x4_f32(
                false, a, false, bfrag[kk], (short)0, acc, false, false);
        }

        // 3) u update + u^2, butterfly-reduce over the 16-lane half per row
#pragma unroll
        for (int v = 0; v < 8; ++v) {
            float un = u[v] + DT_OVER_TAU * (acc[v] + Ie[v] - u[v]);
            u[v] = un;
            float up = fmaxf(un, 0.0f);
            usq[v] = up * up;
            float ps = usq[v];
            ps += __shfl_xor(ps, 1, 32);
            ps += __shfl_xor(ps, 2, 32);
            ps += __shfl_xor(ps, 4, 32);
            ps += __shfl_xor(ps, 8, 32);
            if (l16 == 0) atomicAdd(&rowsum[s & 1][8 * half + v], ps);
        }
        __syncthreads();                              // barrier B

        // 4) divisive norm + STP dynamics (x, su use OLD r)
#pragma unroll
        for (int v = 0; v < 8; ++v) {
            float norm = 1.0f + K_RHO * rowsum[s & 1][8 * half + v];
            float rn = usq[v] / norm;
            float ro = r[v], xo = x[v], so = su[v];
            float dx  = (1.0f - xo) * INV_TAU_D - so * xo * ro;
            float xn  = xo + dx * DT_SEC;
            x[v] = fminf(fmaxf(xn, 0.0f), 1.0f);
            float dsu = (U_STP_C - so) * INV_TAU_F + U_STP_C * (1.0f - so) * ro;
            float sn  = so + dsu * DT_SEC;
            su[v] = fminf(fmaxf(sn, 0.0f), 1.0f);
            r[v] = rn;
        }
    }

    // ---- write back [4, B, N] ----
    if (valid) {
#pragma unroll
        for (int v = 0; v < 8; ++v) {
            int m = v + 8 * half;
            int g = (rowbase + m) * N_REAL + col;
            out[0 * (128 * N_REAL) + g] = u[v];
            out[1 * (128 * N_REAL) + g] = r[v];
            out[2 * (128 * N_REAL) + g] = x[v];
            out[3 * (128 * N_REAL) + g] = su[v];
        }
    }
}

extern "C" void kernel_launch(void* const* d_in, const int* in_sizes, int n_in,
                              void* d_out, int out_size, void* d_ws, size_t ws_size,
                              hipStream_t stream) {
    const float* u    = (const float*)d_in[0];
    const float* r    = (const float*)d_in[1];
    const float* sx   = (const float*)d_in[2];
    const float* suu  = (const float*)d_in[3];
    const float* Iext = (const float*)d_in[4];
    const float* kern = (const float*)d_in[5];
    // d_in[6] = n_steps (256, compile-time constant here)
    float* out = (float*)d_out;

    cann_persistent<<<128 / BROWS, BLOCKT, 0, stream>>>(u, r, sx, suu, Iext, kern, out);
}